// ExactInvertibleVADProjection_44255343018598
// MI455X (gfx1250) — compile-verified
//
#include <hip/hip_runtime.h>

// ---------------------------------------------------------------------------
// out[b, o] = sum_d W[labels[b], o, d] * vad[b, d]
// Reformulated as out(B x 768) = X(B x 32) @ Wt(32 x 768) with
//   X[b, 3c+d]  = (labels[b] == c) ? vad[b,d] : 0      (K padded 30 -> 32)
//   Wt[3c+d, o] = W[c, o, d]
// and computed with chained V_WMMA_F32_16X16X4_F32 (K = 32 = 8 x 4).
// HBM-write-bound: ~403 MB of output -> non-temporal b32 stores.
// ---------------------------------------------------------------------------

typedef __attribute__((ext_vector_type(2))) float v2f;
typedef __attribute__((ext_vector_type(8))) float v8f;

#define B_TOTAL      131072
#define NUM_CLASSES  10
#define OUT_DIM      768
#define IN_DIM       3
#define K_REAL       30                      // NUM_CLASSES * IN_DIM
#define NT_PER_WAVE  4                       // 4 N-tiles of 16 -> 64 cols/wave
#define N_STRIPS     (OUT_DIM / (16 * NT_PER_WAVE))   // 12
#define WAVES_PER_WG 8                       // 256 threads, wave32
#define NUM_WGS      768
#define TOTAL_WAVES  (NUM_WGS * WAVES_PER_WG)         // 6144
#define M_PART       (TOTAL_WAVES / N_STRIPS)         // 512 sample partitions
#define M_GROUPS     (B_TOTAL / 16)                   // 8192 groups of 16 rows

// One element of the expanded one-hot activation X[row, k].
__device__ __forceinline__ float xval(int k, int lab, float vx, float vy, float vz) {
    const int c = k / IN_DIM;            // magic-mul by compiler
    const int d = k - c * IN_DIM;
    const float v = (d == 0) ? vx : ((d == 1) ? vy : vz);
    return (k < K_REAL && lab == c) ? v : 0.0f;
}

__global__ __launch_bounds__(256, 1)
void vad_expert_proj_wmma(const float* __restrict__ vad,
                          const int*   __restrict__ labels,
                          const float* __restrict__ W,
                          float*       __restrict__ out) {
    const int tid  = threadIdx.x;
    const int lane = tid & 31;
    const int half = lane >> 4;          // K-pair selector per ISA VGPR layout
    const int nlan = lane & 15;
    const int wave = blockIdx.x * WAVES_PER_WG + (tid >> 5);

    const int strip  = wave % N_STRIPS;  // which 64-column strip of out
    const int mpart  = wave / N_STRIPS;  // which slice of the 8192 row-groups
    const int n_base = strip * (16 * NT_PER_WAVE);

    // ---- One-time: B fragments (Wt, 32 x 64 strip) into registers.
    // Layout (32-bit B, 4x16): lanes 0-15 hold K = {4kc, 4kc+1},
    // lanes 16-31 hold K = {4kc+2, 4kc+3}; N = lane & 15.
    v2f bf[NT_PER_WAVE][8];
#pragma unroll
    for (int t = 0; t < NT_PER_WAVE; ++t) {
        const int o = n_base + t * 16 + nlan;
#pragma unroll
        for (int kc = 0; kc < 8; ++kc) {
            const int k0 = kc * 4 + half * 2;
            float x0 = 0.0f, x1 = 0.0f;
            if (k0 < K_REAL)
                x0 = W[(k0 / IN_DIM) * (OUT_DIM * IN_DIM) + o * IN_DIM + (k0 % IN_DIM)];
            if (k0 + 1 < K_REAL)
                x1 = W[((k0 + 1) / IN_DIM) * (OUT_DIM * IN_DIM) + o * IN_DIM + ((k0 + 1) % IN_DIM)];
            bf[t][kc].x = x0;
            bf[t][kc].y = x1;
        }
    }

    // ---- Stream over 16-row sample groups.
    for (int g = mpart; g < M_GROUPS; g += M_PART) {
        const int m0  = g * 16;
        const int row = m0 + nlan;       // both lane halves read same 16 rows

        const int   lab = labels[row];
        const float vx  = vad[row * 3 + 0];
        const float vy  = vad[row * 3 + 1];
        const float vz  = vad[row * 3 + 2];

        // Prefetch next group's activations (speculative -> OOB-safe).
        __builtin_prefetch(&labels[row + M_PART * 16], 0, 0);
        __builtin_prefetch(&vad[(row + M_PART * 16) * 3], 0, 0);

        // A fragments (16 x 32 one-hot-expanded activations), ISA layout:
        // lanes 0-15: K = {4kc, 4kc+1}; lanes 16-31: K = {4kc+2, 4kc+3}.
        v2f af[8];
#pragma unroll
        for (int kc = 0; kc < 8; ++kc) {
            const int k0 = kc * 4 + half * 2;
            af[kc].x = xval(k0,     lab, vx, vy, vz);
            af[kc].y = xval(k0 + 1, lab, vx, vy, vz);
        }

#pragma unroll
        for (int t = 0; t < NT_PER_WAVE; ++t) {
            v8f acc = {};
#pragma unroll
            for (int kc = 0; kc < 8; ++kc) {
                acc = __builtin_amdgcn_wmma_f32_16x16x4_f32(
                    /*neg_a=*/false, af[kc],
                    /*neg_b=*/false, bf[t][kc],
                    /*c_mod=*/(short)0, acc,
                    /*reuse_a=*/false, /*reuse_b=*/false);
            }
            // D layout: VGPR r, lanes 0-15 -> (M=r, N=lane); lanes 16-31 ->
            // (M=r+8, N=lane-16). Streaming output: non-temporal stores.
            float* p = out + (size_t)(m0 + half * 8) * OUT_DIM + n_base + t * 16 + nlan;
#pragma unroll
            for (int r = 0; r < 8; ++r)
                __builtin_nontemporal_store(acc[r], p + (size_t)r * OUT_DIM);
        }
    }
}

extern "C" void kernel_launch(void* const* d_in, const int* in_sizes, int n_in,
                              void* d_out, int out_size, void* d_ws, size_t ws_size,
                              hipStream_t stream) {
    const float* vad    = (const float*)d_in[0];
    const int*   labels = (const int*)d_in[1];
    const float* W      = (const float*)d_in[2];
    float*       out    = (float*)d_out;
    (void)in_sizes; (void)n_in; (void)out_size; (void)d_ws; (void)ws_size;

    vad_expert_proj_wmma<<<NUM_WGS, 256, 0, stream>>>(vad, labels, W, out);
}